// KgAdapterSentRGAT_71442486002206
// MI455X (gfx1250) — compile-verified
//
#include <hip/hip_runtime.h>

#define N_NODES 50000
#define E_EDGES 800000
#define HID 96
#define HEADS 4
#define DHEAD 24
#define RELS 38
#define NEG_SLOPE 0.2f

typedef __attribute__((ext_vector_type(2))) float v2f;
typedef __attribute__((ext_vector_type(8))) float v8f;

// ---------------- init: zero output accumulator + softmax state ----------------
__global__ void k_init(float* __restrict__ acc, unsigned* __restrict__ lmaxu,
                       float* __restrict__ denom) {
    const long stride = (long)gridDim.x * blockDim.x;
    const long tid = (long)blockIdx.x * blockDim.x + threadIdx.x;
    for (long i = tid; i < (long)N_NODES * HID; i += stride) acc[i] = 0.0f;
    for (long i = tid; i < (long)N_NODES * HEADS; i += stride) {
        lmaxu[i] = 0u;          // maps below mapped(-inf) -> acts as -infinity
        denom[i] = 0.0f;
    }
}

// ---------------- GEMM h = x @ W via V_WMMA_F32_16X16X4_F32 ----------------
// One wave per 16-row tile; 6 column tiles of 16 (HID=96); K in 24 steps of 4.
__global__ void k_gemm_wmma(const float* __restrict__ x, const float* __restrict__ W,
                            float* __restrict__ h) {
    const int wave = blockIdx.x * (blockDim.x >> 5) + (threadIdx.x >> 5);
    if (wave >= N_NODES / 16) return;           // whole-wave uniform exit
    const int lane = threadIdx.x & 31;
    const int half = lane >> 4;                 // 0: K pair {0,1}, 1: K pair {2,3}
    const int l16  = lane & 15;
    const int row0 = wave * 16;

    v8f acc[6];
#pragma unroll
    for (int c = 0; c < 6; ++c)
#pragma unroll
        for (int j = 0; j < 8; ++j) acc[c][j] = 0.0f;

    // A fragment base: row (row0+l16), K offset half*2  (8-byte aligned)
    const float* xrow = x + (long)(row0 + l16) * HID + half * 2;

    for (int k = 0; k < HID; k += 4) {
        v2f a = *(const v2f*)(xrow + k);        // A: (M=l16, K=k+2*half .. +1)
#pragma unroll
        for (int c = 0; c < 6; ++c) {
            v2f b;                              // B: (K, N=c*16+l16)
            b.x = W[(long)(k + half * 2 + 0) * HID + c * 16 + l16];
            b.y = W[(long)(k + half * 2 + 1) * HID + c * 16 + l16];
            acc[c] = __builtin_amdgcn_wmma_f32_16x16x4_f32(
                false, a, false, b, (short)0, acc[c], false, false);
        }
    }

    // C/D layout: VGPR j -> row j (lanes 0-15) / row j+8 (lanes 16-31), col = lane&15
#pragma unroll
    for (int c = 0; c < 6; ++c)
#pragma unroll
        for (int j = 0; j < 8; ++j)
            h[(long)(row0 + j + half * 8) * HID + c * 16 + l16] = acc[c][j];
}

// ---------------- per-relation attention scores (38 x 4) ----------------
__global__ void k_rel_scores(const float* __restrict__ rel_emb,
                             const float* __restrict__ att_rel,
                             float* __restrict__ s_rel) {
    int i = blockIdx.x * blockDim.x + threadIdx.x;
    if (i >= RELS * HEADS) return;
    int rr = i / HEADS, hh = i % HEADS;
    float s = 0.0f;
#pragma unroll
    for (int d = 0; d < DHEAD; ++d)
        s += rel_emb[(long)rr * HID + hh * DHEAD + d] * att_rel[hh * DHEAD + d];
    s_rel[i] = s;
}

// ---------------- per-node attention scores (N x 4, src & dst) ----------------
__global__ void k_node_scores(const float* __restrict__ h,
                              const float* __restrict__ att_src,
                              const float* __restrict__ att_dst,
                              float* __restrict__ s_src, float* __restrict__ s_dst) {
    int i = blockIdx.x * blockDim.x + threadIdx.x;
    if (i >= N_NODES * HEADS) return;
    int n = i / HEADS, hh = i % HEADS;
    const float* hp = h + (long)n * HID + hh * DHEAD;
    float a = 0.0f, b = 0.0f;
#pragma unroll
    for (int d = 0; d < DHEAD; ++d) {
        float v = hp[d];
        a += v * att_src[hh * DHEAD + d];
        b += v * att_dst[hh * DHEAD + d];
    }
    s_src[i] = a;
    s_dst[i] = b;
}

__device__ __forceinline__ float edge_logit(const float* s_src, const float* s_dst,
                                            const float* s_rel, int s, int d, int t, int hh) {
    float l = s_src[s * HEADS + hh] + s_dst[d * HEADS + hh] + s_rel[t * HEADS + hh];
    return (l > 0.0f) ? l : NEG_SLOPE * l;       // leaky_relu
}

__device__ __forceinline__ unsigned fmap(float f) {   // monotonic float -> uint
    unsigned u = __float_as_uint(f);
    return (u >> 31) ? ~u : (u | 0x80000000u);
}
__device__ __forceinline__ float funmap(unsigned u) {
    return (u >> 31) ? __uint_as_float(u ^ 0x80000000u) : __uint_as_float(~u);
}

// ---------------- segment max of logits per (dst, head) ----------------
__global__ void k_edge_max(const int* __restrict__ ei, const int* __restrict__ et,
                           const float* __restrict__ s_src, const float* __restrict__ s_dst,
                           const float* __restrict__ s_rel, unsigned* __restrict__ lmaxu) {
    int e = blockIdx.x * blockDim.x + threadIdx.x;
    if (e >= E_EDGES) return;
    int s = ei[e], d = ei[E_EDGES + e], t = et[e];
#pragma unroll
    for (int hh = 0; hh < HEADS; ++hh) {
        float l = edge_logit(s_src, s_dst, s_rel, s, d, t, hh);
        atomicMax(&lmaxu[d * HEADS + hh], fmap(l));
    }
}

// ---------------- exp(logit - max), store numerator, sum denominator ----------------
__global__ void k_edge_exp(const int* __restrict__ ei, const int* __restrict__ et,
                           const float* __restrict__ s_src, const float* __restrict__ s_dst,
                           const float* __restrict__ s_rel, const unsigned* __restrict__ lmaxu,
                           float* __restrict__ alpha, float* __restrict__ denom) {
    int e = blockIdx.x * blockDim.x + threadIdx.x;
    if (e >= E_EDGES) return;
    int s = ei[e], d = ei[E_EDGES + e], t = et[e];
#pragma unroll
    for (int hh = 0; hh < HEADS; ++hh) {
        float l = edge_logit(s_src, s_dst, s_rel, s, d, t, hh);
        float lm = funmap(lmaxu[d * HEADS + hh]);
        float ev = __expf(l - lm);
        alpha[(long)e * HEADS + hh] = ev;
        atomicAdd(&denom[d * HEADS + hh], ev);
    }
}

// ---------------- normalize alpha in place ----------------
__global__ void k_edge_norm(const int* __restrict__ ei, const float* __restrict__ denom,
                            float* __restrict__ alpha) {
    long i = (long)blockIdx.x * blockDim.x + threadIdx.x;
    if (i >= (long)E_EDGES * HEADS) return;
    int e = (int)(i >> 2), hh = (int)(i & 3);
    int d = ei[E_EDGES + e];
    alpha[i] = alpha[i] / denom[d * HEADS + hh];
}

// ---------------- weighted message scatter: acc[dst] += alpha*(h[src]+rel) ----------------
__global__ void k_scatter(const int* __restrict__ ei, const int* __restrict__ et,
                          const float* __restrict__ h, const float* __restrict__ rel_emb,
                          const float* __restrict__ alpha, float* __restrict__ acc) {
    long idx = (long)blockIdx.x * blockDim.x + threadIdx.x;
    if (idx >= (long)E_EDGES * HID) return;
    int e  = (int)(idx / HID);
    int dd = (int)(idx % HID);
    int hh = dd / DHEAD;
    int s = ei[e], d = ei[E_EDGES + e], t = et[e];
    float a = alpha[(long)e * HEADS + hh];
    float val = a * (h[(long)s * HID + dd] + rel_emb[(long)t * HID + dd]);
    atomicAdd(&acc[(long)d * HID + dd], val);
}

// ---------------- finalize: out = gelu(acc + bias) (tanh approx, jax default) ----------------
__global__ void k_finalize(const float* __restrict__ bias, float* __restrict__ out) {
    long i = (long)blockIdx.x * blockDim.x + threadIdx.x;
    if (i >= (long)N_NODES * HID) return;
    int c = (int)(i % HID);
    float v = out[i] + bias[c];
    float u = 0.7978845608028654f * (v + 0.044715f * v * v * v);
    out[i] = 0.5f * v * (1.0f + tanhf(u));
}

extern "C" void kernel_launch(void* const* d_in, const int* in_sizes, int n_in,
                              void* d_out, int out_size, void* d_ws, size_t ws_size,
                              hipStream_t stream) {
    const float* x       = (const float*)d_in[0];   // [N, HID]
    const int*   ei      = (const int*)  d_in[1];   // [2, E]
    const int*   et      = (const int*)  d_in[2];   // [E]
    const float* W       = (const float*)d_in[3];   // [HID, HID]
    const float* rel_emb = (const float*)d_in[4];   // [R, HID]
    const float* att_src = (const float*)d_in[5];   // [HEADS, D]
    const float* att_dst = (const float*)d_in[6];
    const float* att_rel = (const float*)d_in[7];
    const float* bias    = (const float*)d_in[8];   // [HID]

    float* out   = (float*)d_out;                   // [N, HID]
    float* alpha = out + (long)N_NODES * HID;       // [E, HEADS]

    // workspace layout (floats)
    float*    ws    = (float*)d_ws;
    float*    h     = ws;                           // N*HID
    float*    s_src = h + (long)N_NODES * HID;      // N*HEADS
    float*    s_dst = s_src + (long)N_NODES * HEADS;
    float*    s_rel = s_dst + (long)N_NODES * HEADS; // 256 slots
    unsigned* lmaxu = (unsigned*)(s_rel + 256);      // N*HEADS
    float*    denom = (float*)(lmaxu + (long)N_NODES * HEADS);

    const int B = 256;

    k_init<<<2048, B, 0, stream>>>(out, lmaxu, denom);

    const int row_tiles = N_NODES / 16;             // 3125
    const int waves_per_block = B / 32;             // 8
    k_gemm_wmma<<<(row_tiles + waves_per_block - 1) / waves_per_block, B, 0, stream>>>(x, W, h);

    k_rel_scores<<<(RELS * HEADS + B - 1) / B, B, 0, stream>>>(rel_emb, att_rel, s_rel);
    k_node_scores<<<(N_NODES * HEADS + B - 1) / B, B, 0, stream>>>(h, att_src, att_dst, s_src, s_dst);

    k_edge_max<<<(E_EDGES + B - 1) / B, B, 0, stream>>>(ei, et, s_src, s_dst, s_rel, lmaxu);
    k_edge_exp<<<(E_EDGES + B - 1) / B, B, 0, stream>>>(ei, et, s_src, s_dst, s_rel, lmaxu, alpha, denom);
    k_edge_norm<<<(int)(((long)E_EDGES * HEADS + B - 1) / B), B, 0, stream>>>(ei, denom, alpha);

    k_scatter<<<(int)(((long)E_EDGES * HID + B - 1) / B), B, 0, stream>>>(ei, et, h, rel_emb, alpha, out);

    k_finalize<<<(int)(((long)N_NODES * HID + B - 1) / B), B, 0, stream>>>(bias, out);
}